// KDNet_17377437679771
// MI455X (gfx1250) — compile-verified
//
#include <hip/hip_runtime.h>
#include <math.h>

// ---------------------------------------------------------------------------
// KD-Net forward for MI455X (gfx1250, wave32, WMMA, async global->LDS).
// Internal activation layout: (N, C, B) so per-point GEMMs over the batch
// have unit-stride columns and layer l output == layer l+1 input.
// Branch-select is applied to *weight rows* before the GEMM (sel depends only
// on the point index), cutting matmul FLOPs 3x vs computing all branches.
// ---------------------------------------------------------------------------

typedef float v2f __attribute__((ext_vector_type(2)));
typedef float v8f __attribute__((ext_vector_type(8)));
typedef int   kdv4i __attribute__((vector_size(16)));   // matches builtin param

#define BATCH 1024

#if defined(__HIP_DEVICE_COMPILE__) && \
    __has_builtin(__builtin_amdgcn_global_load_async_to_lds_b128)
#define KD_HAVE_ASYNC_LDS 1
#else
#define KD_HAVE_ASYNC_LDS 0
#endif

__device__ __forceinline__ void kd_wait_async0()
{
#if KD_HAVE_ASYNC_LDS
#if __has_builtin(__builtin_amdgcn_s_wait_asynccnt)
    __builtin_amdgcn_s_wait_asynccnt(0);
#else
    asm volatile("s_wait_asynccnt 0x0" ::: "memory");
#endif
#endif
}

// ---- transpose (B,3,2048) -> (2048,3,B) -----------------------------------
__global__ void transpose_in_kernel(const float* __restrict__ in,
                                    float* __restrict__ out)
{
    __shared__ float tile[3][32][33];
    const int nb = blockIdx.x;   // 64 tiles over n (2048)
    const int bb = blockIdx.y;   // 32 tiles over b (1024)
    const int tx = threadIdx.x;  // 32
    const int ty = threadIdx.y;  // 8
    for (int c = 0; c < 3; ++c)
        for (int r = ty; r < 32; r += 8)
            tile[c][r][tx] = in[((size_t)(bb * 32 + r) * 3 + c) * 2048 + nb * 32 + tx];
    __syncthreads();
    for (int c = 0; c < 3; ++c)
        for (int r = ty; r < 32; r += 8)
            out[((size_t)(nb * 32 + r) * 3 + c) * 1024 + bb * 32 + tx] = tile[c][tx][r];
}

// ---- fused kdconv layer (templated on compile-time shape) ------------------
// X   : (N, C, B)   input activations
// W   : (3F, C)     weights (row-major)
// bias: (3F)
// sel : (N)
// Y   : (N/2, F, B) output after select + bias + relu + pair-max
//
// Block: 128 threads = 4 waves. blockIdx = (f-tile16, b-tile128, pair p).
// Each wave owns 16f x 32b and both pooled points -> 4 WMMA accumulators.
// LDS: 32 selected weight rows, row-major with +4 float row pad so the 16
// A-fragment lanes hit 16 distinct banks (Cstr % 64 in {8,12,36,4}).
template <int C, int F>
__global__ void __launch_bounds__(128)
kdconv_wmma_kernel(const float* __restrict__ X,
                   const float* __restrict__ W,
                   const float* __restrict__ bias,
                   const int* __restrict__ sel,
                   float* __restrict__ Y)
{
    constexpr int Cpad = (C + 3) & ~3;
    constexpr int Cstr = Cpad + 4;          // pad floats are never read as K
    extern __shared__ float smem[];         // 32 * Cstr floats

    const int f0   = blockIdx.x * 16;
    const int b0   = blockIdx.y * 128;
    const int p    = blockIdx.z;            // max-pool pair index
    const int tid  = threadIdx.x;
    const int lane = tid & 31;
    const int wave = tid >> 5;

    const int s0 = sel[2 * p];
    const int s1 = sel[2 * p + 1];

    // ---- stage the 2x16 selected weight rows into LDS ----
#if KD_HAVE_ASYNC_LDS
    if constexpr ((C & 3) == 0) {
        // gfx1250 async global->LDS copy, 16B chunks, no VGPR round-trip.
        constexpr int CH = C / 4;           // b128 chunks per row
        for (int idx = tid; idx < 32 * CH; idx += 128) {
            const int row = idx / CH;       // compile-time shift (C pow2)
            const int ch  = idx - row * CH;
            const int j   = row >> 4;
            const int r   = row & 15;
            int f = f0 + r;
            if (f > F - 1) f = F - 1;       // dup row; masked at store
            const int s = j ? s1 : s0;
            const float* g = W + (size_t)(f * 3 + s) * C + ch * 4;
            float*       l = smem + (size_t)row * Cstr + ch * 4;
            __builtin_amdgcn_global_load_async_to_lds_b128(
                (__attribute__((address_space(1))) kdv4i*)(g),
                (__attribute__((address_space(3))) kdv4i*)(l), 0, 0);
        }
        kd_wait_async0();
    } else
#endif
    {
        for (int idx = tid; idx < 32 * Cpad; idx += 128) {
            const int row = idx / Cpad;
            const int c   = idx - row * Cpad;
            const int j   = row >> 4;
            const int r   = row & 15;
            int f = f0 + r;
            if (f > F - 1) f = F - 1;
            const int s = j ? s1 : s0;
            smem[(size_t)row * Cstr + c] =
                (c < C) ? W[(size_t)(f * 3 + s) * C + c] : 0.0f;
        }
    }
    __syncthreads();

    const int col   = lane & 15;            // M row for A / batch col in tile
    const int khalf = (lane >> 4) * 2;      // K sub-offset: 0 or 2
    const int bw    = b0 + wave * 32;       // this wave's 32-batch tile

    const float* Xn0 = X + (size_t)(2 * p) * C * BATCH;
    const float* Xn1 = X + (size_t)(2 * p + 1) * C * BATCH;

    v8f acc00 = {};  // point n0, batches [bw, bw+16)
    v8f acc01 = {};  // point n0, batches [bw+16, bw+32)
    v8f acc10 = {};  // point n1, batches [bw, bw+16)
    v8f acc11 = {};  // point n1, batches [bw+16, bw+32)

#pragma unroll 4
    for (int k0 = 0; k0 < Cpad; k0 += 4) {
        const int c0 = k0 + khalf;          // even -> 8B-aligned LDS loads
        const v2f a0 = *(const v2f*)&smem[(size_t)(col)*Cstr + c0];
        const v2f a1 = *(const v2f*)&smem[(size_t)(16 + col) * Cstr + c0];

        v2f b00, b01, b10, b11;
        if constexpr (Cpad == C) {
            b00.x = Xn0[(size_t)(c0    ) * BATCH + bw + col];
            b00.y = Xn0[(size_t)(c0 + 1) * BATCH + bw + col];
            b01.x = Xn0[(size_t)(c0    ) * BATCH + bw + 16 + col];
            b01.y = Xn0[(size_t)(c0 + 1) * BATCH + bw + 16 + col];
            b10.x = Xn1[(size_t)(c0    ) * BATCH + bw + col];
            b10.y = Xn1[(size_t)(c0 + 1) * BATCH + bw + col];
            b11.x = Xn1[(size_t)(c0    ) * BATCH + bw + 16 + col];
            b11.y = Xn1[(size_t)(c0 + 1) * BATCH + bw + 16 + col];
        } else {  // only layer 0 (C=3): guard the K tail
            b00.x = (c0     < C) ? Xn0[(size_t)(c0    ) * BATCH + bw + col] : 0.0f;
            b00.y = (c0 + 1 < C) ? Xn0[(size_t)(c0 + 1) * BATCH + bw + col] : 0.0f;
            b01.x = (c0     < C) ? Xn0[(size_t)(c0    ) * BATCH + bw + 16 + col] : 0.0f;
            b01.y = (c0 + 1 < C) ? Xn0[(size_t)(c0 + 1) * BATCH + bw + 16 + col] : 0.0f;
            b10.x = (c0     < C) ? Xn1[(size_t)(c0    ) * BATCH + bw + col] : 0.0f;
            b10.y = (c0 + 1 < C) ? Xn1[(size_t)(c0 + 1) * BATCH + bw + col] : 0.0f;
            b11.x = (c0     < C) ? Xn1[(size_t)(c0    ) * BATCH + bw + 16 + col] : 0.0f;
            b11.y = (c0 + 1 < C) ? Xn1[(size_t)(c0 + 1) * BATCH + bw + 16 + col] : 0.0f;
        }

        acc00 = __builtin_amdgcn_wmma_f32_16x16x4_f32(false, a0, false, b00,
                                                      (short)0, acc00, false, false);
        acc01 = __builtin_amdgcn_wmma_f32_16x16x4_f32(false, a0, false, b01,
                                                      (short)0, acc01, false, false);
        acc10 = __builtin_amdgcn_wmma_f32_16x16x4_f32(false, a1, false, b10,
                                                      (short)0, acc10, false, false);
        acc11 = __builtin_amdgcn_wmma_f32_16x16x4_f32(false, a1, false, b11,
                                                      (short)0, acc11, false, false);
    }

    // Epilogue: bias + relu + pairwise max fused into the D fragments.
    const int rhalf = (lane >> 4) * 8;
    float* Yp = Y + (size_t)p * F * BATCH;
#pragma unroll
    for (int i = 0; i < 8; ++i) {
        const int f = f0 + i + rhalf;
        if (f < F) {
            const float bb0 = bias[f * 3 + s0];
            const float bb1 = bias[f * 3 + s1];
            const float u0 = fmaxf(acc00[i] + bb0, 0.0f);
            const float u1 = fmaxf(acc10[i] + bb1, 0.0f);
            Yp[(size_t)f * BATCH + bw + col] = fmaxf(u0, u1);
            const float w0 = fmaxf(acc01[i] + bb0, 0.0f);
            const float w1 = fmaxf(acc11[i] + bb1, 0.0f);
            Yp[(size_t)f * BATCH + bw + 16 + col] = fmaxf(w0, w1);
        }
    }
}

// ---- final FC: logits[b,k] = sum_f Xf[f,b] * Wfc[k,f] + bfc[k] -------------
__global__ void __launch_bounds__(32)
fc_wmma_kernel(const float* __restrict__ Xf,    // (1024, B) f-major
               const float* __restrict__ Wfc,   // (16, 1024)
               const float* __restrict__ bfc,   // (16)
               float* __restrict__ logits)      // (B, 16)
{
    const int b0    = blockIdx.x * 16;
    const int lane  = threadIdx.x & 31;
    const int col   = lane & 15;
    const int khalf = (lane >> 4) * 2;

    v8f acc = {};
#pragma unroll 8
    for (int k0 = 0; k0 < 1024; k0 += 4) {
        const int c0 = k0 + khalf;
        v2f a, b;
        a.x = Wfc[(size_t)col * 1024 + c0];
        a.y = Wfc[(size_t)col * 1024 + c0 + 1];
        b.x = Xf[(size_t)(c0    ) * BATCH + b0 + col];
        b.y = Xf[(size_t)(c0 + 1) * BATCH + b0 + col];
        acc = __builtin_amdgcn_wmma_f32_16x16x4_f32(false, a, false, b,
                                                    (short)0, acc, false, false);
    }
    const int rhalf = (lane >> 4) * 8;
#pragma unroll
    for (int i = 0; i < 8; ++i) {
        const int k = i + rhalf;
        logits[(size_t)(b0 + col) * 16 + k] = acc[i] + bfc[k];
    }
}

// ---- log-softmax over 16 classes ------------------------------------------
__global__ void logsoftmax16_kernel(const float* __restrict__ logits,
                                    float* __restrict__ out)
{
    const int b = blockIdx.x * blockDim.x + threadIdx.x;
    if (b >= BATCH) return;
    const float* r = logits + (size_t)b * 16;
    float m = r[0];
    for (int k = 1; k < 16; ++k) m = fmaxf(m, r[k]);
    float s = 0.0f;
    for (int k = 0; k < 16; ++k) s += expf(r[k] - m);
    const float ls = logf(s);
    for (int k = 0; k < 16; ++k) out[(size_t)b * 16 + k] = r[k] - m - ls;
}

// ---------------------------------------------------------------------------
template <int C, int F>
static void launch_layer(const float* Xin, const float* Wl, const float* bl,
                         const int* sl, float* Yout, int N, hipStream_t stream)
{
    constexpr int Cstr = (((C + 3) & ~3)) + 4;
    const size_t shmem = (size_t)32 * Cstr * sizeof(float);
    if (shmem > 65536) {
        // CDNA5 WGPs support up to 320KB LDS per workgroup; raise the cap.
        (void)hipFuncSetAttribute((const void*)kdconv_wmma_kernel<C, F>,
                                  hipFuncAttributeMaxDynamicSharedMemorySize,
                                  (int)shmem);
    }
    dim3 grid((F + 15) / 16, BATCH / 128, N / 2);
    kdconv_wmma_kernel<C, F><<<grid, 128, shmem, stream>>>(Xin, Wl, bl, sl, Yout);
}

extern "C" void kernel_launch(void* const* d_in, const int* in_sizes, int n_in,
                              void* d_out, int out_size, void* d_ws, size_t ws_size,
                              hipStream_t stream)
{
    // dict order: in_x, Ws[0..10], bs[0..10], fc_w, fc_b, sels[0..10]
    const float* in_x = (const float*)d_in[0];
    const float* Ws[11];
    const float* bs[11];
    const int*   sels[11];
    for (int l = 0; l < 11; ++l) Ws[l]   = (const float*)d_in[1 + l];
    for (int l = 0; l < 11; ++l) bs[l]   = (const float*)d_in[12 + l];
    const float* fc_w = (const float*)d_in[23];
    const float* fc_b = (const float*)d_in[24];
    for (int l = 0; l < 11; ++l) sels[l] = (const int*)d_in[25 + l];

    // Workspace: two ping-pong activation buffers (max 67MB each) + logits.
    char* ws = (char*)d_ws;
    float* bufA   = (float*)(ws);
    float* bufB   = (float*)(ws + (size_t)72 * 1024 * 1024);
    float* logits = (float*)(ws + (size_t)144 * 1024 * 1024);

    // (B,3,2048) -> internal (2048,3,B)
    transpose_in_kernel<<<dim3(64, 32), dim3(32, 8), 0, stream>>>(in_x, bufA);

    float* cur = bufA;
    float* nxt = bufB;

#define KD_LAYER(l, Cc, Ff, Nn)                                               \
    do {                                                                      \
        launch_layer<Cc, Ff>(cur, Ws[l], bs[l], sels[l], nxt, Nn, stream);    \
        float* t_ = cur; cur = nxt; nxt = t_;                                 \
    } while (0)

    KD_LAYER(0,   3,    8, 2048);
    KD_LAYER(1,   8,   32, 1024);
    KD_LAYER(2,  32,   64,  512);
    KD_LAYER(3,  64,   64,  256);
    KD_LAYER(4,  64,   64,  128);
    KD_LAYER(5,  64,  128,   64);
    KD_LAYER(6, 128,  256,   32);
    KD_LAYER(7, 256,  512,   16);
    KD_LAYER(8, 512,  512,    8);
    KD_LAYER(9, 512,  512,    4);
    KD_LAYER(10, 512, 1024,   2);
#undef KD_LAYER

    // cur: (1,1024,B) == features f-major
    fc_wmma_kernel<<<BATCH / 16, 32, 0, stream>>>(cur, fc_w, fc_b, logits);
    logsoftmax16_kernel<<<(BATCH + 127) / 128, 128, 0, stream>>>(logits,
                                                                 (float*)d_out);
}